// Router_26645977105051
// MI455X (gfx1250) — compile-verified
//
#include <hip/hip_runtime.h>
#include <hip/hip_bf16.h>
#include <math.h>

// MoE router for MI455X (gfx1250, wave32).
// GEMM [16384x4096] x [4096x8]^T via V_WMMA_F32_16X16X4_F32, then per-token
// softmax / entropy / top-2 epilogue. Bandwidth-bound: 256MB of x @ 23.3TB/s
// => ~11us floor. A-loads are non-temporal (streamed once); gate_w (128KB)
// stays hot in WGP$/L2. B columns 8-15 are harmless duplicates of experts
// 0-7 (never read from C), so all loads are unconditional -> no exec dance,
// deep load batching per unroll-8 block.

typedef __attribute__((ext_vector_type(2))) float v2f;
typedef __attribute__((ext_vector_type(8))) float v8f;

#define RH 4096            // hidden dim
#define RE 8               // experts
#define RT 16384           // tokens
#define WAVES_PER_BLOCK 4

__global__ __launch_bounds__(32 * WAVES_PER_BLOCK)
void router_wmma_kernel(const float* __restrict__ x,
                        const float* __restrict__ gate_w,
                        float* __restrict__ out)
{
    __shared__ float logits_lds[WAVES_PER_BLOCK * 16 * RE];

    const int lane = threadIdx.x & 31;
    const int wave = threadIdx.x >> 5;
    const int tile = blockIdx.x * WAVES_PER_BLOCK + wave; // 16-token tile id
    const int tok0 = tile * 16;

    // fp32 WMMA fragment mapping (ISA 7.12.2):
    //   A 16x4: lane L -> row M = L&15, K pair base = 2*(L>>4)
    //   B 4x16: lane L -> col N = L&15, K pair base = 2*(L>>4)
    const int m  = lane & 15;
    const int kb = (lane >> 4) << 1;

    const float* ap = x      + (size_t)(tok0 + m)  * RH + kb;
    // Unconditional for ALL lanes: cols 8-15 of C become duplicates of
    // experts 0-7 and are never read. No exec masking in the hot loop.
    const float* bp = gate_w + (size_t)(m & (RE-1)) * RH + kb;

    v8f c = {0.f, 0.f, 0.f, 0.f, 0.f, 0.f, 0.f, 0.f};

    // K loop: 1024 WMMA steps of K=4, unrolled x8 -> 16 loads in flight.
    #pragma unroll 8
    for (int k = 0; k < RH; k += 4) {
        v2f a = __builtin_nontemporal_load((const v2f*)(ap + k)); // stream x (NT)
        v2f b = *(const v2f*)(bp + k);                            // cached gate
        c = __builtin_amdgcn_wmma_f32_16x16x4_f32(
                /*neg_a=*/false, a, /*neg_b=*/false, b,
                /*c_mod=*/(short)0, c, /*reuse_a=*/false, /*reuse_b=*/false);
    }

    // C layout: lanes 0-15: VGPR r -> M=r ; lanes 16-31: VGPR r -> M=r+8 ; N = lane&15.
    float* lrow = logits_lds + wave * (16 * RE);
    if (m < RE) {
        const int trow = (lane >> 4) * 8;
        #pragma unroll
        for (int r = 0; r < 8; ++r)
            lrow[(trow + r) * RE + m] = c[r];
    }
    __syncthreads();

    // Epilogue: lanes 0..15 each own one token of this wave's tile.
    if (lane < 16) {
        const int t = tok0 + lane;
        const float* l = lrow + lane * RE;

        float lv[RE];
        float mx = -INFINITY;
        #pragma unroll
        for (int e = 0; e < RE; ++e) { lv[e] = l[e]; mx = fmaxf(mx, lv[e]); }

        float p[RE];
        float s = 0.f;
        #pragma unroll
        for (int e = 0; e < RE; ++e) { p[e] = expf(lv[e] - mx); s += p[e]; }
        const float inv = 1.f / s;

        float ent = 0.f;
        #pragma unroll
        for (int e = 0; e < RE; ++e) { p[e] *= inv; ent -= p[e] * logf(p[e] + 1e-9f); }

        // top-2, ties -> lowest index first (strict > while scanning in order)
        int i0 = 0; float w0 = p[0];
        #pragma unroll
        for (int e = 1; e < RE; ++e) if (p[e] > w0) { w0 = p[e]; i0 = e; }
        int i1 = (i0 == 0) ? 1 : 0; float w1 = p[i1];
        #pragma unroll
        for (int e = 0; e < RE; ++e) if (e != i0 && p[e] > w1) { w1 = p[e]; i1 = e; }
        const float rn = 1.f / (w0 + w1 + 1e-9f);

        // concatenated outputs (all float32 in d_out), reference return order
        float* topw   = out;                 // [T,2]
        float* topi   = out + 2 * RT;        // [T,2] indices as float
        float* probs  = out + 4 * RT;        // [T,8]
        float* probsc = out + 12 * RT;       // [T,8]
        float* logit  = out + 20 * RT;       // [T,8]
        float* entp   = out + 28 * RT;       // [T]

        topw[t * 2 + 0] = w0 * rn;
        topw[t * 2 + 1] = w1 * rn;
        topi[t * 2 + 0] = (float)i0;
        topi[t * 2 + 1] = (float)i1;
        #pragma unroll
        for (int e = 0; e < RE; ++e) {
            probs [t * RE + e] = p[e];
            probsc[t * RE + e] = p[e];
            logit [t * RE + e] = lv[e];
        }
        entp[t] = ent;
    }
}

extern "C" void kernel_launch(void* const* d_in, const int* in_sizes, int n_in,
                              void* d_out, int out_size, void* d_ws, size_t ws_size,
                              hipStream_t stream) {
    const float* x      = (const float*)d_in[0];
    const float* gate_w = (const float*)d_in[1];
    float* out          = (float*)d_out;

    const int tiles  = RT / 16;                       // 1024 waves
    const int blocks = tiles / WAVES_PER_BLOCK;       // 256 blocks x 128 threads
    router_wmma_kernel<<<blocks, 32 * WAVES_PER_BLOCK, 0, stream>>>(x, gate_w, out);
}